// InterpolativeUpsamplerCUDA_1039382086123
// MI455X (gfx1250) — compile-verified
//
#include <hip/hip_runtime.h>
#include <stdint.h>

// upsample2d (up=2, 5x5 FIR, gain=4): x=(2048 planes,128,128) f32 -> (2048,256,256) f32.
// Bandwidth-bound (0.67 GB @ 23.3 TB/s ~= 29 us floor; compute ~1.7 GFLOP is negligible).
// CDNA5 path: TDM tensor_load_to_lds (iterate mode, 1 row/iteration) stages the haloed
// input tile into LDS — right halo zero-filled by TDM OOB semantics, top/left/bottom
// pre-zeroed (disjoint, race-free) — then s_wait_tensorcnt + barrier, polyphase 2x2
// compute, and non-temporal B64 stores for the streamed output.

#define HH 128                // plane height (reference-fixed)
#define WW 128                // plane width  (reference-fixed)
#define TILE_H 32
#define TILE_W 64
#define HALO_W (TILE_W + 2)   // 66
#define HALO_H (TILE_H + 2)   // 34
#define LDS_STRIDE 68
#define NTHREADS 256          // 8 wave32 waves
#define TILES_PER_PLANE 8     // (128/32) * (128/64), blockIdx: [plane:11][ty:2][tx:1]

typedef float v2f  __attribute__((ext_vector_type(2)));
typedef unsigned int u32x4 __attribute__((ext_vector_type(4)));
typedef int i32x4 __attribute__((ext_vector_type(4)));
typedef int i32x8 __attribute__((ext_vector_type(8)));

#if defined(__has_builtin)
#if __has_builtin(__builtin_amdgcn_tensor_load_to_lds) && __has_builtin(__builtin_amdgcn_s_wait_tensorcnt)
#define USE_TDM 1
#endif
#endif
#ifndef USE_TDM
#define USE_TDM 0
#endif

__global__ __launch_bounds__(NTHREADS) void upsample2d_tdm_kernel(
    const float* __restrict__ x, const float* __restrict__ kern,
    float* __restrict__ out)
{
  __shared__ float lds[HALO_H * LDS_STRIDE];

  const int b     = blockIdx.x;
  const int plane = b >> 3;              // TILES_PER_PLANE == 8, no division
  const int ty0   = ((b >> 1) & 3) * TILE_H;
  const int tx0   = (b & 1) * TILE_W;

  const float* __restrict__ xp = x + (size_t)plane * (HH * WW);
  float* __restrict__ op = out + (size_t)plane * (4 * HH * WW);

  const int tid = threadIdx.x;

  // ---------------- stage haloed input tile into LDS ----------------
#if USE_TDM
  {
    const int rowshift = (ty0 == 0) ? 1 : 0;
    const int colshift = (tx0 == 0) ? 1 : 0;
    const int gy_start = ty0 - 1 + rowshift;
    const int gx_start = tx0 - 1 + colshift;
    const int nrows = min(HALO_H - rowshift, HH - gy_start);
    const int ncols = HALO_W - colshift;   // right-edge OOB zero-filled by TDM

    // Pre-zero only the LDS cells the TDM will NOT write (disjoint -> no race).
    if (rowshift) { for (int c = tid; c < HALO_W; c += NTHREADS) lds[c] = 0.f; }
    const int last = rowshift + nrows;     // first row the TDM does not write
    if (last < HALO_H) { for (int c = tid; c < HALO_W; c += NTHREADS) lds[last * LDS_STRIDE + c] = 0.f; }
    if (colshift) { for (int r = tid; r < HALO_H; r += NTHREADS) lds[r * LDS_STRIDE] = 0.f; }
    __syncthreads();

    if (tid == 0) {
      const uint64_t ga    = (uint64_t)(uintptr_t)(xp + (size_t)gy_start * WW + gx_start);
      const uint32_t lbase = (uint32_t)(uintptr_t)(&lds[0]);
      const uint32_t laddr = lbase + (uint32_t)(rowshift * LDS_STRIDE + colshift) * 4u;

      // D# group 0: count=1, lds_addr, global_addr, type=2 (image)
      u32x4 g0;
      g0.x = 1u;
      g0.y = laddr;
      g0.z = (uint32_t)(ga & 0xffffffffu);
      g0.w = (uint32_t)((ga >> 32) & 0x01ffffffu) | 0x80000000u;

      const uint32_t td0   = (uint32_t)(WW - gx_start);  // x >= td0 -> zero fill
      const uint32_t td1   = (uint32_t)HH;
      const uint32_t tile0 = (uint32_t)ncols;            // one row per iteration
      const uint64_t s0    = (uint64_t)WW;               // tensor_dim0_stride

      // D# group 1: data_size=4B (2), iterate_enable, dims/tile dims/strides
      i32x8 g1;
      g1[0] = (int)((2u << 16) | (1u << 19));
      g1[1] = (int)((td0 & 0xffffu) << 16);                       // atomic_addr=0 | td0[15:0]
      g1[2] = (int)((td0 >> 16) | ((td1 & 0xffffu) << 16));       // td0[31:16] | td1[15:0]
      g1[3] = (int)((td1 >> 16) | ((tile0 & 0xffffu) << 16));     // td1[31:16] | tile_dim0
      g1[4] = (int)1;                                             // tile_dim1=1, tile_dim2=0
      g1[5] = (int)(uint32_t)(s0 & 0xffffffffu);                  // dim0_stride[31:0]
      g1[6] = (int)(uint32_t)((s0 >> 32) & 0xffffu);              // dim0_stride[47:32]
      g1[7] = 0;

      // D# group 2 (iterate mode): lds_addr_increment, global_addr_increment, count
      const uint64_t gainc = (uint64_t)WW;
      i32x4 g2;
      g2[0] = 1;                                                  // tensor_dim2 (unused)
      g2[1] = LDS_STRIDE;                                         // lds step (elements)
      g2[2] = (int)(uint32_t)(gainc & 0xffffffffu);               // global step (elements)
      g2[3] = (int)((uint32_t)((gainc >> 32) & 0xffffu) | ((uint32_t)(nrows - 1) << 16));
      i32x4 g3 = {0, 0, 0, 0};

      // This toolchain's builtin takes 6 args: one extra (reserved) 8-DWORD group.
      i32x8 g4 = {0, 0, 0, 0, 0, 0, 0, 0};

      __builtin_amdgcn_tensor_load_to_lds(g0, g1, g2, g3, g4, 0);
      __builtin_amdgcn_s_wait_tensorcnt(0);
    }
    __syncthreads();
  }
#else
  for (int idx = tid; idx < HALO_H * HALO_W; idx += NTHREADS) {
    const int r  = idx / HALO_W, c = idx - r * HALO_W;
    const int gy = ty0 - 1 + r,  gx = tx0 - 1 + c;
    float v = 0.f;
    if (gy >= 0 && gy < HH && gx >= 0 && gx < WW) v = xp[(size_t)gy * WW + gx];
    lds[r * LDS_STRIDE + c] = v;
  }
  __syncthreads();
#endif

  // ---------------- per-phase 2D tap weights (uniform, gain=4 folded in) ----------------
  // even output phase: input offsets {-1,0}   <-> flipped-kernel indices {3,1}
  // odd  output phase: input offsets {-1,0,1} <-> indices {4,2,0}
  float P00[2][2], P01[2][3], P10[3][2], P11[3][3];
  {
    const int se[2] = {3, 1};
    const int so[3] = {4, 2, 0};
#pragma unroll
    for (int a = 0; a < 2; ++a)
#pragma unroll
      for (int c = 0; c < 2; ++c) P00[a][c] = 4.f * kern[se[a] * 5 + se[c]];
#pragma unroll
    for (int a = 0; a < 2; ++a)
#pragma unroll
      for (int c = 0; c < 3; ++c) P01[a][c] = 4.f * kern[se[a] * 5 + so[c]];
#pragma unroll
    for (int a = 0; a < 3; ++a)
#pragma unroll
      for (int c = 0; c < 2; ++c) P10[a][c] = 4.f * kern[so[a] * 5 + se[c]];
#pragma unroll
    for (int a = 0; a < 3; ++a)
#pragma unroll
      for (int c = 0; c < 3; ++c) P11[a][c] = 4.f * kern[so[a] * 5 + so[c]];
  }

  // ---------------- compute: 8 positions/thread, 2x2 outputs each ----------------
  const int lj   = tid & (TILE_W - 1);   // 0..63 (lane-consecutive -> coalesced)
  const int lrow = tid >> 6;             // 0..3
  const int W2   = 2 * WW;

#pragma unroll
  for (int p = 0; p < TILE_H / 4; ++p) {
    const int li = lrow + p * 4;
    const float* l0 = &lds[(li + 0) * LDS_STRIDE + lj];
    const float* l1 = &lds[(li + 1) * LDS_STRIDE + lj];
    const float* l2 = &lds[(li + 2) * LDS_STRIDE + lj];
    const float a00 = l0[0], a01 = l0[1], a02 = l0[2];
    const float a10 = l1[0], a11 = l1[1], a12 = l1[2];
    const float a20 = l2[0], a21 = l2[1], a22 = l2[2];

    const float o00 = a00*P00[0][0] + a01*P00[0][1]
                    + a10*P00[1][0] + a11*P00[1][1];
    const float o01 = a00*P01[0][0] + a01*P01[0][1] + a02*P01[0][2]
                    + a10*P01[1][0] + a11*P01[1][1] + a12*P01[1][2];
    const float o10 = a00*P10[0][0] + a01*P10[0][1]
                    + a10*P10[1][0] + a11*P10[1][1]
                    + a20*P10[2][0] + a21*P10[2][1];
    const float o11 = a00*P11[0][0] + a01*P11[0][1] + a02*P11[0][2]
                    + a10*P11[1][0] + a11*P11[1][1] + a12*P11[1][2]
                    + a20*P11[2][0] + a21*P11[2][1] + a22*P11[2][2];

    const size_t obase = (size_t)(2 * (ty0 + li)) * W2 + (size_t)(2 * (tx0 + lj));
    v2f r0 = {o00, o01};
    v2f r1 = {o10, o11};
    __builtin_nontemporal_store(r0, (v2f*)(op + obase));       // streamed output: NT
    __builtin_nontemporal_store(r1, (v2f*)(op + obase + W2));
  }
}

extern "C" void kernel_launch(void* const* d_in, const int* in_sizes, int n_in,
                              void* d_out, int out_size, void* d_ws, size_t ws_size,
                              hipStream_t stream) {
  (void)n_in; (void)out_size; (void)d_ws; (void)ws_size;
  const float* x    = (const float*)d_in[0];
  const float* kern = (const float*)d_in[1];
  float* out        = (float*)d_out;

  const int planes = in_sizes[0] / (HH * WW);    // N*C = 2048
  dim3 grid((unsigned)(planes * TILES_PER_PLANE));
  upsample2d_tdm_kernel<<<grid, NTHREADS, 0, stream>>>(x, kern, out);
}